// TestModel_59201829208124
// MI455X (gfx1250) — compile-verified
//
#include <hip/hip_runtime.h>

typedef __attribute__((ext_vector_type(2))) float v2f;
typedef __attribute__((ext_vector_type(8))) float v8f;

#define N_ROWS 128
#define D_IN   262144
#define D1     64
#define D2     32
#define D3     16
#define KBLK   128                 // split-K blocks
#define CHUNK  (D_IN / KBLK)       // 2048 K per block

// ---------------------------------------------------------------------------
// Kernel 1: split-K GEMM  partial[blk] = x[:, kslab] @ W1[kslab, :]
// One block = 8 waves; wave w owns M-tile w (16 rows), loops the 4 N-tiles
// reusing its A fragment. Uses v_wmma_f32_16x16x4_f32 (fp32 WMMA).
// ---------------------------------------------------------------------------
__global__ __launch_bounds__(256) void gemm1_splitk(
    const float* __restrict__ x, const float* __restrict__ W1,
    float* __restrict__ partial) {
  const int blk  = blockIdx.x;
  const int lane = threadIdx.x & 31;
  const int wave = threadIdx.x >> 5;     // 0..7 -> M tile
  const int half = lane >> 4;            // 0: K even pair, 1: K odd pair
  const int l15  = lane & 15;
  const int m    = wave * 16 + l15;      // A-fragment row (lane = M)
  const size_t kbase = (size_t)blk * CHUNK;

  v8f acc0 = {0.f,0.f,0.f,0.f,0.f,0.f,0.f,0.f};
  v8f acc1 = acc0, acc2 = acc0, acc3 = acc0;

  const float* xptr = x  + (size_t)m * D_IN + kbase + 2 * half;
  const float* wptr = W1 + (kbase + 2 * half) * D1 + l15;

  #pragma unroll 2
  for (int k = 0; k < CHUNK; k += 4) {
    v2f a;                                   // A 16x4: lane holds K=k+2h, k+1+2h
    a.x = xptr[k];
    a.y = xptr[k + 1];
    const float* wb = wptr + (size_t)k * D1; // B 4x16: lane = N, VGPR/half = K
    v2f b;
    b.x = wb[0];        b.y = wb[D1];
    acc0 = __builtin_amdgcn_wmma_f32_16x16x4_f32(false, a, false, b, (short)0, acc0, false, false);
    b.x = wb[16];       b.y = wb[D1 + 16];
    acc1 = __builtin_amdgcn_wmma_f32_16x16x4_f32(false, a, false, b, (short)0, acc1, false, false);
    b.x = wb[32];       b.y = wb[D1 + 32];
    acc2 = __builtin_amdgcn_wmma_f32_16x16x4_f32(false, a, false, b, (short)0, acc2, false, false);
    b.x = wb[48];       b.y = wb[D1 + 48];
    acc3 = __builtin_amdgcn_wmma_f32_16x16x4_f32(false, a, false, b, (short)0, acc3, false, false);
  }

  float* o = partial + (size_t)blk * (N_ROWS * D1);
  #pragma unroll
  for (int r = 0; r < 8; ++r) {            // C/D: row = r + 8*half, col = l15
    const int row = wave * 16 + r + 8 * half;
    o[row * D1 +      l15] = acc0[r];
    o[row * D1 + 16 + l15] = acc1[r];
    o[row * D1 + 32 + l15] = acc2[r];
    o[row * D1 + 48 + l15] = acc3[r];
  }
}

// ---------------------------------------------------------------------------
// Kernel 2: deterministic split-K reduction + bias + relu -> h [128,64]
// ---------------------------------------------------------------------------
__global__ __launch_bounds__(256) void reduce_bias_relu(
    const float* __restrict__ partial, const float* __restrict__ b1,
    float* __restrict__ h) {
  const int idx = blockIdx.x * blockDim.x + threadIdx.x;  // 0..8191
  float s = 0.f;
  #pragma unroll 4
  for (int b = 0; b < KBLK; ++b) s += partial[(size_t)b * (N_ROWS * D1) + idx];
  s += b1[idx & (D1 - 1)];
  h[idx] = fmaxf(s, 0.f);
}

// ---------------------------------------------------------------------------
// Kernel 3 (single block): gather by indices, stable partition by rowsum>0,
// per-row descending sort (== top_k k=64 of 64), then two tiny WMMA GEMMs.
// ---------------------------------------------------------------------------
__global__ __launch_bounds__(256) void tail_kernel(
    const float* __restrict__ h, const int* __restrict__ indices,
    const float* __restrict__ W2, const float* __restrict__ b2,
    const float* __restrict__ W3, const float* __restrict__ b3,
    float* __restrict__ out) {
  __shared__ float sh_h[N_ROWS][D1];
  __shared__ float sh_mid[N_ROWS][D2];
  __shared__ int   sh_cond[N_ROWS];
  __shared__ int   sh_pos[N_ROWS];

  const int tid = threadIdx.x;

  // cond[i] = sum(relu(h[indices[i]])) > 0  (h already >= 0)
  if (tid < N_ROWS) {
    const float* src = h + (size_t)indices[tid] * D1;
    float s = 0.f;
    for (int c = 0; c < D1; ++c) s += src[c];
    sh_cond[tid] = (s > 0.f) ? 1 : 0;
  }
  __syncthreads();

  // stable-partition destination: zeros keep order first, then ones
  if (tid < N_ROWS) {
    const int ci = sh_cond[tid];
    int before_same = 0, zeros_total = 0;
    for (int j = 0; j < N_ROWS; ++j) {
      zeros_total += (sh_cond[j] == 0);
      if (j < tid && sh_cond[j] == ci) ++before_same;
    }
    sh_pos[tid] = (ci == 0) ? before_same : (zeros_total + before_same);
  }
  __syncthreads();

  // scatter gathered rows into partitioned order
  if (tid < N_ROWS) {
    const float* src = h + (size_t)indices[tid] * D1;
    float* dst = &sh_h[sh_pos[tid]][0];
    for (int c = 0; c < D1; ++c) dst[c] = src[c];
  }
  __syncthreads();

  // top_k(k=64) over 64 features == full descending sort of each row
  if (tid < N_ROWS) {
    float* row = &sh_h[tid][0];
    for (int i = 1; i < D1; ++i) {
      float v = row[i];
      int j = i - 1;
      while (j >= 0 && row[j] < v) { row[j + 1] = row[j]; --j; }
      row[j + 1] = v;
    }
  }
  __syncthreads();

  const int lane = tid & 31;
  const int wave = tid >> 5;
  const int half = lane >> 4;
  const int l15  = lane & 15;
  const int m    = wave * 16 + l15;

  // GEMM2: sh_h[128,64] @ W2[64,32] + b2, relu -> sh_mid
  {
    v8f acc0 = {0.f,0.f,0.f,0.f,0.f,0.f,0.f,0.f};
    v8f acc1 = acc0;
    #pragma unroll
    for (int k = 0; k < D1; k += 4) {
      v2f a, b0, b1v;
      a.x   = sh_h[m][k + 2 * half];
      a.y   = sh_h[m][k + 1 + 2 * half];
      b0.x  = W2[(k + 2 * half) * D2 + l15];
      b0.y  = W2[(k + 1 + 2 * half) * D2 + l15];
      b1v.x = W2[(k + 2 * half) * D2 + 16 + l15];
      b1v.y = W2[(k + 1 + 2 * half) * D2 + 16 + l15];
      acc0 = __builtin_amdgcn_wmma_f32_16x16x4_f32(false, a, false, b0,  (short)0, acc0, false, false);
      acc1 = __builtin_amdgcn_wmma_f32_16x16x4_f32(false, a, false, b1v, (short)0, acc1, false, false);
    }
    #pragma unroll
    for (int r = 0; r < 8; ++r) {
      const int row = wave * 16 + r + 8 * half;
      sh_mid[row][l15]      = fmaxf(acc0[r] + b2[l15], 0.f);
      sh_mid[row][16 + l15] = fmaxf(acc1[r] + b2[16 + l15], 0.f);
    }
  }
  __syncthreads();

  // GEMM3: sh_mid[128,32] @ W3[32,16] + b3 -> out[128,16]
  {
    v8f acc = {0.f,0.f,0.f,0.f,0.f,0.f,0.f,0.f};
    #pragma unroll
    for (int k = 0; k < D2; k += 4) {
      v2f a, b;
      a.x = sh_mid[m][k + 2 * half];
      a.y = sh_mid[m][k + 1 + 2 * half];
      b.x = W3[(k + 2 * half) * D3 + l15];
      b.y = W3[(k + 1 + 2 * half) * D3 + l15];
      acc = __builtin_amdgcn_wmma_f32_16x16x4_f32(false, a, false, b, (short)0, acc, false, false);
    }
    #pragma unroll
    for (int r = 0; r < 8; ++r) {
      const int row = wave * 16 + r + 8 * half;
      out[row * D3 + l15] = acc[r] + b3[l15];
    }
  }
}

// ---------------------------------------------------------------------------
extern "C" void kernel_launch(void* const* d_in, const int* in_sizes, int n_in,
                              void* d_out, int out_size, void* d_ws, size_t ws_size,
                              hipStream_t stream) {
  const float* x       = (const float*)d_in[0];
  const int*   indices = (const int*)  d_in[1];
  const float* W1      = (const float*)d_in[2];
  const float* b1      = (const float*)d_in[3];
  const float* W2      = (const float*)d_in[4];
  const float* b2      = (const float*)d_in[5];
  const float* W3      = (const float*)d_in[6];
  const float* b3      = (const float*)d_in[7];
  float* out = (float*)d_out;

  float* partial = (float*)d_ws;                              // KBLK * 8192 f32
  float* h       = partial + (size_t)KBLK * (N_ROWS * D1);    // 8192 f32

  gemm1_splitk<<<KBLK, 256, 0, stream>>>(x, W1, partial);
  reduce_bias_relu<<<(N_ROWS * D1) / 256, 256, 0, stream>>>(partial, b1, h);
  tail_kernel<<<1, 256, 0, stream>>>(h, indices, W2, b2, W3, b3, out);
}